// Attn_mg_59098749993124
// MI455X (gfx1250) — compile-verified
//
#include <hip/hip_runtime.h>
#include <hip/hip_bf16.h>

// ---------------------------------------------------------------------------
// MeshCNN fused block, algebraically folded through the 3-row fuse conv.
//   out[b,r,e] = sum_K A[r,K] * F[K,b,e] + bias[r],   K = 1280
// F layout per path p (K base = p*640):
//   K = p*640 + c*4 + j, j=0..3 : [f1+f3, f2+f4, |f1-f3|, |f2-f4|]  (gathered)
//   K = p*640 + 512 + c         : x[c,e]  (identity slot, absorbs local conv)
// Matrix math: V_WMMA_F32_16X16X4_F32 (fp32-exact; kernel is gather-bound).
// Gather dedup: lanes m / m+16 split the 4 neighbor loads (2 each) and trade
// the missing half via DS_SWIZZLE_B32 SWAPX16 (halves scattered vmem issue).
// ---------------------------------------------------------------------------

typedef float v2f __attribute__((ext_vector_type(2)));
typedef float v8f __attribute__((ext_vector_type(8)));

#define CCH   128
#define EE    60000
#define BATCH 2
#define NTILE (EE / 16)            // 3750 tiles of 16 edges
#define KTOT  1280
#define NKK   (KTOT / 4)           // 320 wmma steps per tile
#define APACK_FLOATS (NKK * 2 * 16 * 2)   // 20480 floats = 80 KB
#define SWZ_SWAPX16 0x401f         // ds_swizzle: xor=0x10, or=0, and=0x1f

// ---------------------------------------------------------------------------
// Phase 1: fold fuse matrices through conv weights, emit A in WMMA lane layout
// Apack[kk][h][m][j] = A[m][kk*4 + h*2 + j]  (h = lane>>4, m = lane&15)
// ---------------------------------------------------------------------------
__global__ void mesh_fold_kernel(
    const float* __restrict__ Wal, const float* __restrict__ bal,
    const float* __restrict__ Wbl, const float* __restrict__ bbl,
    const float* __restrict__ Wat, const float* __restrict__ bat,
    const float* __restrict__ Wbt, const float* __restrict__ bbt,
    const float* __restrict__ Waf, const float* __restrict__ baf,
    const float* __restrict__ Wbf, const float* __restrict__ bbf,
    float* __restrict__ apack)
{
    for (int idx = threadIdx.x; idx < APACK_FLOATS + 16; idx += blockDim.x) {
        float v = 0.0f;
        if (idx < APACK_FLOATS) {
            const int j  = idx & 1;
            const int m  = (idx >> 1) & 15;
            const int h  = (idx >> 5) & 1;
            const int kk = idx >> 6;
            const int K  = kk * 4 + h * 2 + j;
            if (m < 3) {
                const int p  = K / 640;
                const int kp = K % 640;
                const float* Wf = p ? Wbf : Waf;   // [3,256]
                const float* Wt = p ? Wbt : Wat;   // [128,128,5]
                const float* Wl = p ? Wbl : Wal;   // [128,128]
                float acc = 0.0f;
                if (kp < 512) {                    // symmetric-feature slots
                    const int c = kp >> 2;
                    const int s = (kp & 3) + 1;    // s = 1..4
                    for (int o = 0; o < CCH; ++o)
                        acc += Wf[m * 256 + 128 + o] * Wt[(o * CCH + c) * 5 + s];
                } else {                           // identity slot (s=0 + local)
                    const int c = kp - 512;
                    for (int o = 0; o < CCH; ++o)
                        acc += Wf[m * 256 + 128 + o] * Wt[(o * CCH + c) * 5 + 0]
                             + Wf[m * 256 + o]       * Wl[o * CCH + c];
                }
                v = acc;
            }
        } else {                                   // folded bias constant
            const int r = idx - APACK_FLOATS;
            if (r < 3) {
                float acc = baf[r] + bbf[r];
                for (int o = 0; o < CCH; ++o)
                    acc += Waf[r * 256 + o]       * bal[o]
                         + Waf[r * 256 + 128 + o] * bat[o]
                         + Wbf[r * 256 + o]       * bbl[o]
                         + Wbf[r * 256 + 128 + o] * bbt[o];
                v = acc;
            }
        }
        apack[idx] = v;
    }
}

// ---------------------------------------------------------------------------
// Phase 2: streaming pass. One wave per 16-edge tile, 320 fp32 WMMAs/tile.
// ---------------------------------------------------------------------------
__global__ __launch_bounds__(256)
void mesh_fused_main(const float* __restrict__ x0, const float* __restrict__ x1,
                     const int*   __restrict__ gi, const float* __restrict__ apack,
                     float* __restrict__ out)
{
    extern __shared__ float Alds[];                // 20496 floats (80 KB + 64 B)
    for (int i = threadIdx.x; i < APACK_FLOATS + 16; i += 256)
        Alds[i] = apack[i];
    __syncthreads();

    const int lane = threadIdx.x & 31;
    const int wave = threadIdx.x >> 5;
    const int tile = blockIdx.x * 8 + wave;        // wave-uniform
    if (tile >= BATCH * NTILE) return;

    const int b    = tile / NTILE;
    const int m    = lane & 15;                    // N / M position
    const int half = lane >> 4;                    // K sub-slice select
    const int e    = (tile % NTILE) * 16 + m;      // edge owned by this lane

    const int4 nb = *(const int4*)(gi + (((size_t)b * EE) + e) * 4);
    // Gather split: half0 owns the (n1,n3) pair, half1 owns the (n2,n4) pair.
    const int nA = half ? nb.y : nb.x;
    const int nB = half ? nb.w : nb.z;

    const float* xpa = x0 + (size_t)b * CCH * EE;
    const float* xpb = x1 + (size_t)b * CCH * EE;

    v8f acc = {};                                  // 16x16 fp32 accumulator

    for (int p = 0; p < 2; ++p) {
        const float* xb     = p ? xpb : xpa;
        const int    kkbase = p * 160;

        // --- gathered symmetric-feature block: 128 WMMAs -------------------
        // B-operand K layout per channel: (f1+f3, f2+f4 | |f1-f3|, |f2-f4|)
        for (int c = 0; c < CCH; ++c) {
            const float* row = xb + (size_t)c * EE;
            const float gA  = row[nA];
            const float gB  = row[nB];
            const float sum = gA + gB;             // h0: g1+g3   h1: g2+g4
            const float dif = fabsf(gA - gB);      // h0: |g1-g3| h1: |g2-g4|
            // cross-half exchange: h0 sends dif (partner needs |g1-g3|),
            // h1 sends sum (partner needs g2+g4)
            const float send = half ? sum : dif;
            const float recv = __int_as_float(
                __builtin_amdgcn_ds_swizzle(__float_as_int(send), SWZ_SWAPX16));
            v2f bv;                                // branchless: EXEC stays full
            bv.x = half ? recv : sum;              // K = 4c + (0|2)
            bv.y = half ? dif  : recv;             // K = 4c + (1|3)
            const v2f av = *(const v2f*)&Alds[(((kkbase + c) * 2 + half) * 16 + m) * 2];
            acc = __builtin_amdgcn_wmma_f32_16x16x4_f32(
                      false, av, false, bv, (short)0, acc, false, false);
        }

        // --- identity (x) block: 32 WMMAs ----------------------------------
        for (int cc = 0; cc < 32; ++cc) {
            const int c0 = cc * 4 + half * 2;
            v2f bv;
            bv.x = xb[(size_t)c0 * EE + e];
            bv.y = xb[(size_t)(c0 + 1) * EE + e];
            const v2f av = *(const v2f*)&Alds[(((kkbase + 128 + cc) * 2 + half) * 16 + m) * 2];
            acc = __builtin_amdgcn_wmma_f32_16x16x4_f32(
                      false, av, false, bv, (short)0, acc, false, false);
        }
    }

    // D layout: VGPR j, lanes 0-15 hold M=j. Rows 0..2 are the real outputs.
    if (half == 0) {
        out[((size_t)b * 3 + 0) * EE + e] = acc[0] + Alds[APACK_FLOATS + 0];
        out[((size_t)b * 3 + 1) * EE + e] = acc[1] + Alds[APACK_FLOATS + 1];
        out[((size_t)b * 3 + 2) * EE + e] = acc[2] + Alds[APACK_FLOATS + 2];
    }
}

// ---------------------------------------------------------------------------
extern "C" void kernel_launch(void* const* d_in, const int* in_sizes, int n_in,
                              void* d_out, int out_size, void* d_ws, size_t ws_size,
                              hipStream_t stream)
{
    const float* x0  = (const float*)d_in[0];
    const float* x1  = (const float*)d_in[1];
    const int*   gi  = (const int*)  d_in[2];
    const float* Wal = (const float*)d_in[3];
    const float* bal = (const float*)d_in[4];
    const float* Wbl = (const float*)d_in[5];
    const float* bbl = (const float*)d_in[6];
    const float* Wat = (const float*)d_in[7];
    const float* bat = (const float*)d_in[8];
    const float* Wbt = (const float*)d_in[9];
    const float* bbt = (const float*)d_in[10];
    const float* Waf = (const float*)d_in[11];
    const float* baf = (const float*)d_in[12];
    const float* Wbf = (const float*)d_in[13];
    const float* bbf = (const float*)d_in[14];

    float* apack = (float*)d_ws;                  // 20496 floats = ~82 KB

    mesh_fold_kernel<<<1, 256, 0, stream>>>(Wal, bal, Wbl, bbl,
                                            Wat, bat, Wbt, bbt,
                                            Waf, baf, Wbf, bbf, apack);

    const int ntiles = BATCH * NTILE;             // 7500 waves of work
    const dim3 grid((ntiles + 7) / 8);
    const size_t lds = (size_t)(APACK_FLOATS + 16) * sizeof(float);
    mesh_fused_main<<<grid, 256, lds, stream>>>(x0, x1, gi, apack, (float*)d_out);
}